// EntropyBasedPatcher_82927228551670
// MI455X (gfx1250) — compile-verified
//
#include <hip/hip_runtime.h>

// ---- problem constants (from reference) ----
#define BB 8
#define SS 16384
#define WIN 8
#define MINP 4
#define MAXP 32
#define THRV 2.0f
#define NP (SS / MINP)            // 4096
#define SEARCH_LEN (MAXP - MINP)  // 28

// ---- flat output layout (floats, reference return order) ----
#define OFF_PATCH 0
#define OFF_LEN   (BB * NP * MAXP)            // 1048576
#define OFF_POS   (OFF_LEN + BB * NP)         // 1081344
#define OFF_WENT  (OFF_POS + BB * NP * 2)     // 1146880  (total 1277952)

// =====================================================================
// Kernel 1: sliding-window byte entropy, window=8, 256-symbol alphabet.
// One thread per (b, s). Tail positions replicate ent[S-WIN].
// ent = sum over 256 symbols of -(c/8+1e-10)*log2(c/8+1e-10); only 9
// distinct per-symbol terms exist (c in 0..8), computed once per block.
// The count-class histogram makes the result independent of byte
// arrangement, so equal multisets tie bit-exactly (argmin-stable).
// =====================================================================
__global__ void __launch_bounds__(256)
entropy_kernel(const int* __restrict__ seq,
               const float* __restrict__ wgt,
               float* __restrict__ ent_ws,     // raw entropy [B*S]
               float* __restrict__ went_out) { // weighted entropy [B*S]
    __shared__ float s_term[WIN + 1];
    if (threadIdx.x <= WIN) {
        float p = (float)threadIdx.x * (1.0f / (float)WIN) + 1e-10f;
        s_term[threadIdx.x] = -(p * log2f(p));
    }
    __syncthreads();

    const int gid = blockIdx.x * blockDim.x + threadIdx.x;
    if (gid >= BB * SS) return;
    const int b = gid / SS;
    const int s = gid - b * SS;
    const int sw = (s > (SS - WIN)) ? (SS - WIN) : s;  // tail replication
    const int* w = seq + b * SS + sw;

    int by[WIN];
#pragma unroll
    for (int j = 0; j < WIN; ++j) by[j] = w[j];

    // per-position match counts c_j = #{k : b_k == b_j}
    int c[WIN];
#pragma unroll
    for (int j = 0; j < WIN; ++j) {
        int cj = 0;
#pragma unroll
        for (int k = 0; k < WIN; ++k) cj += (by[j] == by[k]) ? 1 : 0;
        c[j] = cj;
    }

    // arrangement-independent accumulation over count classes
    float ent = 0.0f;
    int ndist = 0;
#pragma unroll
    for (int cc = 1; cc <= WIN; ++cc) {
        int m = 0;
#pragma unroll
        for (int j = 0; j < WIN; ++j) m += (c[j] == cc) ? 1 : 0;
        const int n = m / cc;  // exact: m is a multiple of cc
        ndist += n;
        ent += (float)n * s_term[cc];
    }
    ent += (float)(256 - ndist) * s_term[0];

    ent_ws[gid]   = ent;
    went_out[gid] = wgt[0] * ent;
}

// =====================================================================
// Kernel 2: greedy entropy segmentation. One workgroup per batch row.
// 80 KB LDS/workgroup (within the 320 KB WGP budget): the entropy row
// (64 KB) is staged with gfx1250 async global->LDS DMA (ASYNCcnt), the
// byte row is packed int32->u8 (16 KB). Wave 0 runs the sequential
// greedy scan over the ACTIVE prefix only (each step strictly advances
// start, so active slots == prefix); per step, lanes 0..27 hold the
// search window and argmin = 5x fminf shuffle + ballot + ctz
// (first-index tie-break, matching jnp.argmin). All 8 waves then
// zero-fill the inactive tail with vectorized stores.
// =====================================================================
__global__ void __launch_bounds__(256)
segment_kernel(const int* __restrict__ seq,
               const float* __restrict__ ent_ws,
               float* __restrict__ out) {
    __shared__ float         s_ent[SS];   // 64 KB
    __shared__ unsigned char s_seq[SS];   // 16 KB
    __shared__ int           s_nact;

    const int b = blockIdx.x;
    const int t = threadIdx.x;

    // ---- stage entropy row: 4096 x b128 async chunks, 16 per thread ----
    {
        const float* g = ent_ws + b * SS;
#pragma unroll
        for (int r = 0; r < 16; ++r) {
            const int k = r * 256 + t;            // b128 chunk id
            const float* gp = g + k * 4;
#if defined(__gfx1250__)
            // low 32 bits of a generic pointer to LDS == LDS byte offset
            unsigned ldsoff = (unsigned)(unsigned long long)(&s_ent[k * 4]);
            asm volatile("global_load_async_to_lds_b128 %0, %1, off"
                         :: "v"(ldsoff), "v"(gp) : "memory");
#else
            *(float4*)(&s_ent[k * 4]) = *(const float4*)gp;
#endif
        }
    }
    // ---- stage byte row: int32 -> packed u8 (VALU path, overlaps DMA) ----
    {
        const int4* g4 = (const int4*)(seq + b * SS);
        unsigned* s32 = (unsigned*)s_seq;
#pragma unroll
        for (int r = 0; r < 16; ++r) {
            const int k = r * 256 + t;            // int4 id (4096 total)
            const int4 v = g4[k];
            const unsigned packed = (unsigned)(v.x & 255)
                                  | ((unsigned)(v.y & 255) << 8)
                                  | ((unsigned)(v.z & 255) << 16)
                                  | ((unsigned)(v.w & 255) << 24);
            s32[k] = packed;
        }
    }
#if defined(__gfx1250__)
    asm volatile("s_wait_asynccnt 0x0" ::: "memory");  // async DMA -> LDS done
#endif
    __syncthreads();

    float* patches   = out + OFF_PATCH + (size_t)b * NP * MAXP;
    float* lengths   = out + OFF_LEN   + (size_t)b * NP;
    float* positions = out + OFF_POS   + (size_t)b * NP * 2;

    // ---- sequential greedy scan over the active prefix (wave 0) ----
    if (t < 32) {
        const int lane = t;
        int start = 0;
        int p = 0;
        while (p < NP && start < SS) {
            const int end0 = (start + MAXP < SS) ? (start + MAXP) : SS;
            const int ss   = start + MINP;
            // search window values (masked to +inf like the reference)
            float v = __int_as_float(0x7f800000);
            if (lane < SEARCH_LEN && (ss + lane) < end0) v = s_ent[ss + lane];
            // wave32 min-reduce (no NaNs possible: entropy >= 0 or +inf)
            float minv = v;
#pragma unroll
            for (int off = 16; off > 0; off >>= 1)
                minv = fminf(minv, __shfl_xor(minv, off, 32));
            // first-index tie-break via ballot (lowest set bit)
#if __has_builtin(__builtin_amdgcn_ballot_w32)
            unsigned mask = __builtin_amdgcn_ballot_w32(v == minv);
#else
            unsigned mask = (unsigned)__ballot(v == minv);
#endif
            const int idx = __builtin_ctz(mask);
            // do_split == (end0<S) & (end0-start>MINP) & (ss<se) & (min<THR)
            // which reduces to:
            const bool do_split = (start + MAXP < SS) && (minv < THRV);
            const int end    = do_split ? (ss + idx + 1) : end0;
            const int length = end - start;

            float pv = 0.0f;
            if (lane < length) pv = (float)s_seq[start + lane];
            patches[(size_t)p * MAXP + lane] = pv;
            if (lane == 0) {
                lengths[p] = (float)length;
                *(float2*)(positions + (size_t)p * 2) =
                    make_float2((float)start, (float)end);
            }
            start = end;
            ++p;
        }
        if (lane == 0) s_nact = p;
    }
    __syncthreads();

    // ---- cooperative zero-fill of the inactive tail (all 8 waves) ----
    const int nact = s_nact;
    {   // patches tail: float4 stores (128B-aligned region start)
        float4* p4 = (float4*)patches;
        const float4 z4 = make_float4(0.f, 0.f, 0.f, 0.f);
        for (int i = nact * (MAXP / 4) + t; i < NP * (MAXP / 4); i += 256)
            p4[i] = z4;
    }
    for (int i = nact + t; i < NP; i += 256)
        lengths[i] = 0.0f;
    {   // positions tail: float2 stores (8B-aligned region start)
        float2* p2 = (float2*)positions;
        for (int i = nact + t; i < NP; i += 256)
            p2[i] = make_float2(0.f, 0.f);
    }
}

// =====================================================================
extern "C" void kernel_launch(void* const* d_in, const int* in_sizes, int n_in,
                              void* d_out, int out_size, void* d_ws, size_t ws_size,
                              hipStream_t stream) {
    (void)in_sizes; (void)n_in; (void)out_size; (void)ws_size;
    const int*   seq = (const int*)d_in[0];
    const float* w   = (const float*)d_in[1];
    float* out    = (float*)d_out;
    float* ent_ws = (float*)d_ws;  // B*S floats = 512 KB scratch

    entropy_kernel<<<dim3((BB * SS + 255) / 256), dim3(256), 0, stream>>>(
        seq, w, ent_ws, out + OFF_WENT);
    segment_kernel<<<dim3(BB), dim3(256), 0, stream>>>(seq, ent_ws, out);
}